// TransformerBlock_987842478213
// MI455X (gfx1250) — compile-verified
//
#include <hip/hip_runtime.h>
#include <hip/hip_bf16.h>
#include <math.h>

#define NQ 16384
#define NKV 4096
#define DD 8
#define HH 64

typedef __attribute__((ext_vector_type(2))) float v2f;
typedef __attribute__((ext_vector_type(8))) float v8f;

__device__ __forceinline__ void layernorm8(const float* t, const float* w,
                                           const float* b, float* o) {
  float mu = 0.f;
#pragma unroll
  for (int i = 0; i < 8; ++i) mu += t[i];
  mu *= 0.125f;
  float var = 0.f;
#pragma unroll
  for (int i = 0; i < 8; ++i) { float d = t[i] - mu; var += d * d; }
  var *= 0.125f;
  float rs = rsqrtf(var + 1e-5f);
#pragma unroll
  for (int i = 0; i < 8; ++i) o[i] = (t[i] - mu) * rs * w[i] + b[i];
}

// ---- Kernel 1: LN(y) -> K = w@Wk^T, V = w@Wv^T ------------------------------
__global__ __launch_bounds__(256) void prep_y_kernel(
    const float* __restrict__ y, const float* __restrict__ Wk,
    const float* __restrict__ Wv, const float* __restrict__ lnw,
    const float* __restrict__ lnb, float* __restrict__ K, float* __restrict__ V,
    int m) {
  __shared__ float sWk[64], sWv[64], slw[8], slb[8];
  int t = threadIdx.x;
  if (t < 64) { sWk[t] = Wk[t]; sWv[t] = Wv[t]; }
  if (t < 8) { slw[t] = lnw[t]; slb[t] = lnb[t]; }
  __syncthreads();
  int row = blockIdx.x * blockDim.x + t;
  if (row >= m) return;
  float tr[8], w[8];
#pragma unroll
  for (int i = 0; i < 8; ++i) tr[i] = y[row * 8 + i];
  layernorm8(tr, slw, slb, w);
#pragma unroll
  for (int o = 0; o < 8; ++o) {
    float ak = 0.f, av = 0.f;
#pragma unroll
    for (int i = 0; i < 8; ++i) {
      ak = fmaf(w[i], sWk[o * 8 + i], ak);
      av = fmaf(w[i], sWv[o * 8 + i], av);
    }
    K[row * 8 + o] = ak;
    V[row * 8 + o] = av;
  }
}

// ---- Kernel 2: LN(x) -> Q = z@Wq^T * (1/sqrt(8)) ----------------------------
__global__ __launch_bounds__(256) void prep_x_kernel(
    const float* __restrict__ x, const float* __restrict__ Wq,
    const float* __restrict__ lnw, const float* __restrict__ lnb,
    float* __restrict__ Q, int n) {
  __shared__ float sWq[64], slw[8], slb[8];
  int t = threadIdx.x;
  if (t < 64) sWq[t] = Wq[t];
  if (t < 8) { slw[t] = lnw[t]; slb[t] = lnb[t]; }
  __syncthreads();
  int row = blockIdx.x * blockDim.x + t;
  if (row >= n) return;
  float tr[8], z[8];
#pragma unroll
  for (int i = 0; i < 8; ++i) tr[i] = x[row * 8 + i];
  layernorm8(tr, slw, slb, z);
  const float scale = 0.35355339059327373f;  // 1/sqrt(8)
#pragma unroll
  for (int o = 0; o < 8; ++o) {
    float a = 0.f;
#pragma unroll
    for (int i = 0; i < 8; ++i) a = fmaf(z[i], sWq[o * 8 + i], a);
    Q[row * 8 + o] = a * scale;
  }
}

// ---- Kernel 3: fused smooth-softmax attention -------------------------------
// One wave handles a 16-query tile over all 4096 keys.
// Q.K^T via v_wmma_f32_16x16x4_f32 (two chained K=4 steps for D=8).
__global__ __launch_bounds__(256) void attn_kernel(
    const float* __restrict__ Q, const float* __restrict__ K,
    const float* __restrict__ V, const float* __restrict__ x,
    float* __restrict__ Z1) {
  const int lane = threadIdx.x & 31;
  const int wave = threadIdx.x >> 5;
  const int lx = lane & 15;
  const int half = lane >> 4;
  const int m0 = (blockIdx.x * 8 + wave) * 16;

  // A-fragment (16x4 f32 layout): lane lx = M row, half selects K pair.
  const float* qbase = Q + (m0 + lx) * 8 + 2 * half;
  v2f a0 = *(const v2f*)(qbase);       // d = 2*half + {0,1}
  v2f a1 = *(const v2f*)(qbase + 4);   // d = 4 + 2*half + {0,1}

  float accS[8][8], accR[8][8], sumS[8], sumR[8];
#pragma unroll
  for (int g = 0; g < 8; ++g) {
    sumS[g] = 0.f;
    sumR[g] = 0.f;
#pragma unroll
    for (int d = 0; d < 8; ++d) { accS[g][d] = 0.f; accR[g][d] = 0.f; }
  }

  for (int j0 = 0; j0 < NKV; j0 += 16) {
    // B-fragment (4x16 f32 layout): lane lx = N col (key), half selects K pair.
    const float* kbase = K + (j0 + lx) * 8 + 2 * half;
    v2f b0 = *(const v2f*)(kbase);
    v2f b1 = *(const v2f*)(kbase + 4);
    v8f c = {};
    c = __builtin_amdgcn_wmma_f32_16x16x4_f32(false, a0, false, b0, (short)0, c,
                                              false, false);
    c = __builtin_amdgcn_wmma_f32_16x16x4_f32(false, a1, false, b1, (short)0, c,
                                              false, false);
    // This lane's key row of V (lanes 16-31 duplicate lanes 0-15; L1 hit).
    const float4 vlo = *(const float4*)(V + (j0 + lx) * 8);
    const float4 vhi = *(const float4*)(V + (j0 + lx) * 8 + 4);
    float vrow[8] = {vlo.x, vlo.y, vlo.z, vlo.w, vhi.x, vhi.y, vhi.z, vhi.w};
#pragma unroll
    for (int g = 0; g < 8; ++g) {
      // C layout: vgpr g, lane -> row m0 + g + 8*half, col j0 + lx
      float s = c[g];
      // scores are O(1) by construction; hardware exp (v_exp_f32) is accurate
      // to ~1 ulp here and both softmax numerator and denominator use it.
      float e = __expf(s);
      float r = fmaxf(s, 0.f);
      sumS[g] += e;
      sumR[g] += r;
#pragma unroll
      for (int d = 0; d < 8; ++d) {
        accS[g][d] = fmaf(e, vrow[d], accS[g][d]);
        accR[g][d] = fmaf(r, vrow[d], accR[g][d]);
      }
    }
  }

  // 1) Reduce the row sums across the 16 lanes of each half
  //    (xor masks < 16 keep the two halves independent in wave32).
#pragma unroll
  for (int g = 0; g < 8; ++g) {
#pragma unroll
    for (int msk = 1; msk < 16; msk <<= 1) {
      sumS[g] += __shfl_xor(sumS[g], msk, 32);
      sumR[g] += __shfl_xor(sumR[g], msk, 32);
    }
  }

  // 2) Apply the (row-uniform) normalization to per-lane partials, then reduce
  //    only the combined accumulator -> half the shuffle traffic.
  float comb[8][8];
#pragma unroll
  for (int g = 0; g < 8; ++g) {
    float inv_s = 1.0f / sumS[g];
    float inv_d = 1.0f / (0.1f * sumR[g] + 1.0f);
#pragma unroll
    for (int d = 0; d < 8; ++d)
      comb[g][d] = (0.1f * accR[g][d] + accS[g][d] * inv_s) * inv_d;
  }
#pragma unroll
  for (int g = 0; g < 8; ++g)
#pragma unroll
    for (int msk = 1; msk < 16; msk <<= 1)
#pragma unroll
      for (int d = 0; d < 8; ++d)
        comb[g][d] += __shfl_xor(comb[g][d], msk, 32);

  if (lx == 0) {
#pragma unroll
    for (int g = 0; g < 8; ++g) {
      int row = m0 + g + 8 * half;
      float ov[8];
#pragma unroll
      for (int d = 0; d < 8; ++d) ov[d] = comb[g][d] + x[row * 8 + d];
      *(float4*)(Z1 + row * 8) = make_float4(ov[0], ov[1], ov[2], ov[3]);
      *(float4*)(Z1 + row * 8 + 4) = make_float4(ov[4], ov[5], ov[6], ov[7]);
    }
  }
}

// ---- Kernel 4: LN -> MLP(8->64->64->8) + residual -> out --------------------
__global__ __launch_bounds__(256) void mlp_kernel(
    const float* __restrict__ Z1, const float* __restrict__ W1,
    const float* __restrict__ b1, const float* __restrict__ W2,
    const float* __restrict__ b2, const float* __restrict__ W3,
    const float* __restrict__ b3, const float* __restrict__ lnw,
    const float* __restrict__ lnb, float* __restrict__ out, int n) {
  __shared__ float sW1[512], sW2[4096], sW3[512], sb1[64], sb2[64], sb3[8],
      slw[8], slb[8];
  int t = threadIdx.x;
  for (int i = t; i < 512; i += 256) { sW1[i] = W1[i]; sW3[i] = W3[i]; }
  for (int i = t; i < 4096; i += 256) sW2[i] = W2[i];
  if (t < 64) { sb1[t] = b1[t]; sb2[t] = b2[t]; }
  if (t < 8) { sb3[t] = b3[t]; slw[t] = lnw[t]; slb[t] = lnb[t]; }
  __syncthreads();
  int row = blockIdx.x * blockDim.x + t;
  if (row >= n) return;
  float z[8], zn[8];
#pragma unroll
  for (int i = 0; i < 8; ++i) z[i] = Z1[row * 8 + i];
  layernorm8(z, slw, slb, zn);
  float h1[64];
  for (int o = 0; o < 64; ++o) {
    float a = sb1[o];
#pragma unroll
    for (int i = 0; i < 8; ++i) a = fmaf(zn[i], sW1[o * 8 + i], a);
    h1[o] = fmaxf(a, 0.f);
  }
  float h2[64];
  for (int o = 0; o < 64; ++o) {
    float a = sb2[o];
#pragma unroll 8
    for (int i = 0; i < 64; ++i) a = fmaf(h1[i], sW2[o * 64 + i], a);
    h2[o] = fmaxf(a, 0.f);
  }
#pragma unroll
  for (int o = 0; o < 8; ++o) {
    float a = sb3[o];
#pragma unroll 8
    for (int i = 0; i < 64; ++i) a = fmaf(h2[i], sW3[o * 64 + i], a);
    out[row * 8 + o] = a + z[o];
  }
}

extern "C" void kernel_launch(void* const* d_in, const int* in_sizes, int n_in,
                              void* d_out, int out_size, void* d_ws,
                              size_t ws_size, hipStream_t stream) {
  const float* x = (const float*)d_in[0];
  const float* y = (const float*)d_in[1];
  const float* Wq = (const float*)d_in[2];
  const float* Wk = (const float*)d_in[3];
  const float* Wv = (const float*)d_in[4];
  const float* W1 = (const float*)d_in[5];
  const float* b1 = (const float*)d_in[6];
  const float* W2 = (const float*)d_in[7];
  const float* b2 = (const float*)d_in[8];
  const float* W3 = (const float*)d_in[9];
  const float* b3 = (const float*)d_in[10];
  const float* lnw = (const float*)d_in[11];
  const float* lnb = (const float*)d_in[12];
  float* outp = (float*)d_out;

  const int n = in_sizes[0] / DD;  // 16384
  const int m = in_sizes[1] / DD;  // 4096

  float* ws = (float*)d_ws;
  float* Qw = ws;                  // n*8
  float* Kw = Qw + (size_t)n * 8;  // m*8
  float* Vw = Kw + (size_t)m * 8;  // m*8
  float* Z1 = Vw + (size_t)m * 8;  // n*8

  prep_x_kernel<<<(n + 255) / 256, 256, 0, stream>>>(x, Wq, lnw, lnb, Qw, n);
  prep_y_kernel<<<(m + 255) / 256, 256, 0, stream>>>(y, Wk, Wv, lnw, lnb, Kw,
                                                     Vw, m);
  // one wave per 16 queries, 8 waves per block
  attn_kernel<<<n / (16 * 8), 256, 0, stream>>>(Qw, Kw, Vw, x, Z1);
  mlp_kernel<<<(n + 255) / 256, 256, 0, stream>>>(Z1, W1, b1, W2, b2, W3, b3,
                                                  lnw, lnb, outp, n);
}